// LocalMixer_47579647705675
// MI455X (gfx1250) — compile-verified
//
#include <hip/hip_runtime.h>

// FP32 WMMA local mixer for MI455X (gfx1250, wave32).
// out[b,h,t*16+go] = sum_gi W[h,go,gi] * x[b,h,t*16+gi]
// => per (b,h): Y(16x16) = X(16x16, rows=t, cols=g) * W[h]^T
// => four chained V_WMMA_F32_16X16X4_F32 (K-slices of 4 over g).

typedef __attribute__((ext_vector_type(2))) float v2f;
typedef __attribute__((ext_vector_type(8))) float v8f;

#define HIDDEN 512
#define BATCH  256
#define SEQ    256
#define NB     8      // batch tiles per wave (W[h] cached in registers across them)

__global__ __launch_bounds__(256) void local_mixer_wmma_kernel(
    const float* __restrict__ x,      // (B, HIDDEN, 1, SEQ) f32
    const float* __restrict__ w,      // (HIDDEN, 16, 16)    f32
    float* __restrict__ out)          // (B, HIDDEN, 1, SEQ) f32
{
    const int lane = threadIdx.x & 31;
    const int wave = threadIdx.x >> 5;
    const int widx = blockIdx.x * 8 + wave;   // global wave id: 0 .. 16383
    const int h    = widx & (HIDDEN - 1);     // consecutive waves -> consecutive h
    const int bgrp = widx >> 9;               // 0 .. 31

    const int row  = lane & 15;               // M-row (A) / N-col (B) owned by lane
    const int half = lane >> 4;               // K-pair selector (ISA 16x4 f32 layout)

    // Per ISA 7.12.2 (32-bit A 16x4): lane l, VGPR v holds A[l&15, 2*(l>>4)+v].
    // B (4x16) mirrors with lane = N. For both X and W[h] tiles this is the
    // same float index: row*16 + 4*k + 2*half, an 8-byte-aligned b64 load.
    const int lidx = row * 16 + 2 * half;

    // B operand = W[h]^T slices, loaded once per wave, reused for NB batch tiles.
    const float* wp = w + (size_t)h * 256;
    const v2f b0 = *(const v2f*)(wp + lidx + 0);
    const v2f b1 = *(const v2f*)(wp + lidx + 4);
    const v2f b2 = *(const v2f*)(wp + lidx + 8);
    const v2f b3 = *(const v2f*)(wp + lidx + 12);

#pragma unroll
    for (int i = 0; i < NB; ++i) {
        const int b = bgrp * NB + i;
        const size_t tile = ((size_t)b * HIDDEN + (size_t)h) * SEQ;

        const float* xp = x + tile;
        v2f a0 = *(const v2f*)(xp + lidx + 0);
        v2f a1 = *(const v2f*)(xp + lidx + 4);
        v2f a2 = *(const v2f*)(xp + lidx + 8);
        v2f a3 = *(const v2f*)(xp + lidx + 12);

        v8f c = {};
        // (neg_a, A, neg_b, B, c_mod, C, reuse_a, reuse_b)
        c = __builtin_amdgcn_wmma_f32_16x16x4_f32(false, a0, false, b0, (short)0, c, false, false);
        c = __builtin_amdgcn_wmma_f32_16x16x4_f32(false, a1, false, b1, (short)0, c, false, false);
        c = __builtin_amdgcn_wmma_f32_16x16x4_f32(false, a2, false, b2, (short)0, c, false, false);
        c = __builtin_amdgcn_wmma_f32_16x16x4_f32(false, a3, false, b3, (short)0, c, false, false);

        // D layout: VGPR j, lane l -> Y[j + 8*half, row]. Each store writes two
        // full contiguous 64B lines (lanes 0-15 row j, lanes 16-31 row j+8).
        float* op = out + tile;
#pragma unroll
        for (int j = 0; j < 8; ++j)
            op[(size_t)(j + 8 * half) * 16 + row] = c[j];
    }
}

extern "C" void kernel_launch(void* const* d_in, const int* in_sizes, int n_in,
                              void* d_out, int out_size, void* d_ws, size_t ws_size,
                              hipStream_t stream) {
    const float* x = (const float*)d_in[0];   // 256*512*1*256 f32
    const float* w = (const float*)d_in[1];   // 512*16*16 f32
    float* out = (float*)d_out;               // 256*512*1*256 f32

    // total waves = BATCH/NB * HIDDEN = 32 * 512 = 16384; 8 waves per 256-thread block
    const int blocks = (BATCH / NB) * HIDDEN / 8;   // 2048
    local_mixer_wmma_kernel<<<blocks, 256, 0, stream>>>(x, w, out);
}